// MultiScaleTokenization_37907381354596
// MI455X (gfx1250) — compile-verified
//
#include <hip/hip_runtime.h>

typedef __attribute__((ext_vector_type(2))) float v2f;
typedef __attribute__((ext_vector_type(8))) float v8f;

#define BB     8
#define CC     128
#define HWP    262144      // 512*512
#define NSC    4           // scales
#define NSEG_  1024
#define EMB    192
#define ROWS   1025        // NSEG + cls

// ---------------------------------------------------------------------------
// Zero the workspace (sums + counts) every call (atomic accumulation target).
// ---------------------------------------------------------------------------
__global__ void zero_ws(float* __restrict__ p, int n) {
    int i = blockIdx.x * blockDim.x + threadIdx.x;
    int stride = gridDim.x * blockDim.x;
    for (; i < n; i += stride) p[i] = 0.0f;
}

// ---------------------------------------------------------------------------
// Segment reduction with LDS-privatized histograms.
// grid = (4 pixel tiles, 16 channel chunks, 8 batches), block = 256 (8 waves).
// Dynamic LDS: 8ch * 4scales * 1024segs sums (128KB) + 4*1024 counts (16KB).
// Layout lds[((c*4+s)*1024 + seg)] -> bank = seg % 64 (conflict-friendly).
// ---------------------------------------------------------------------------
__global__ void __launch_bounds__(256) seg_reduce(
        const float* __restrict__ x, const int* __restrict__ seg,
        float* __restrict__ sums, float* __restrict__ cnts) {
    extern __shared__ float lds[];
    const int tid   = threadIdx.x;
    const int tile  = blockIdx.x;   // pixel quarter
    const int chunk = blockIdx.y;   // 8-channel chunk
    const int b     = blockIdx.z;

    for (int i = tid; i < 36864; i += 256) lds[i] = 0.0f;
    __syncthreads();

    const int p0 = tile * 65536;
    const float* xc = x + (size_t)(b * CC + chunk * 8) * HWP;
    const int*   sg = seg + (size_t)b * NSC * HWP;

    for (int it = 0; it < 256; ++it) {
        const int p = p0 + it * 256 + tid;
        const int s0 = sg[0 * HWP + p];
        const int s1 = sg[1 * HWP + p];
        const int s2 = sg[2 * HWP + p];
        const int s3 = sg[3 * HWP + p];
        #pragma unroll
        for (int c = 0; c < 8; ++c) {
            const float v = xc[(size_t)c * HWP + p];
            atomicAdd(&lds[(c * 4 + 0) * 1024 + s0], v);
            atomicAdd(&lds[(c * 4 + 1) * 1024 + s1], v);
            atomicAdd(&lds[(c * 4 + 2) * 1024 + s2], v);
            atomicAdd(&lds[(c * 4 + 3) * 1024 + s3], v);
        }
        if (chunk == 0) {   // counts computed once (independent of channels)
            atomicAdd(&lds[32768 + 0 * 1024 + s0], 1.0f);
            atomicAdd(&lds[32768 + 1 * 1024 + s1], 1.0f);
            atomicAdd(&lds[32768 + 2 * 1024 + s2], 1.0f);
            atomicAdd(&lds[32768 + 3 * 1024 + s3], 1.0f);
        }
    }
    __syncthreads();

    // Flush partials to global (sums laid out [scale][b*NSEG+g][C]).
    for (int i = tid; i < 32768; i += 256) {
        const int c = i >> 12, rem = i & 4095, s = rem >> 10, g = rem & 1023;
        const float v = lds[i];
        if (v != 0.0f)
            atomicAdd(&sums[((size_t)(s * 8192 + b * 1024 + g)) * CC + chunk * 8 + c], v);
    }
    if (chunk == 0) {
        for (int i = tid; i < 4096; i += 256) {
            const int s = i >> 10, g = i & 1023;
            const float v = lds[32768 + i];
            if (v != 0.0f)
                atomicAdd(&cnts[s * 8192 + b * 1024 + g], v);
        }
    }
}

// ---------------------------------------------------------------------------
// Token GEMM via V_WMMA_F32_16X16X4_F32.
// One wave computes a 16x48 strip (3 N-tiles share one A fragment).
// Branchless A construction: a[kk] = p[kk]*s1 + q[kk]*s2 covers cls rows,
// mean rows (sums * 1/cnt) and zero-padded rows with one uniform expression,
// so the unrolled K loop has no control flow and loads can be clause-batched.
// ---------------------------------------------------------------------------
__global__ void __launch_bounds__(256) token_gemm(
        const float* __restrict__ sums, const float* __restrict__ cnts,
        const float* __restrict__ cls_tok, const float* __restrict__ cls_pos,
        const float* __restrict__ w, const float* __restrict__ bias,
        float* __restrict__ out) {
    const int lane = threadIdx.x & 31;
    const int wave = threadIdx.x >> 5;
    const int l15  = lane & 15;
    const int half = lane >> 4;

    int t = blockIdx.x * 8 + wave;          // 8320 wave-strips, exact fit
    const int ng = t % 4;   t /= 4;         // N group of 48
    const int mt = t % 65;  t /= 65;
    const int b  = t % 8;
    const int s  = t / 8;
    const int m0 = mt * 16, n0 = ng * 48;

    // Per-lane A row (ISA 16x4 f32 A layout: M = lane%16, K pair = 2*half).
    const int arow = m0 + l15;              // 0..1039
    const float* p;
    const float* q = cls_pos;
    float s1, s2;
    if (arow == 0) {                        // cls row: cls_token + cls_pos
        p = cls_tok; s1 = 1.0f; s2 = 1.0f;
    } else if (arow < ROWS) {               // mean row: sums * 1/max(cnt,1)
        const int g = s * 8192 + b * 1024 + (arow - 1);
        p  = sums + (size_t)g * CC;
        s1 = 1.0f / fmaxf(cnts[g], 1.0f);
        s2 = 0.0f;
    } else {                                // padded row: zero
        p = sums; s1 = 0.0f; s2 = 0.0f;
    }

    const float* wp0 = w + (size_t)(n0 + l15)      * CC;
    const float* wp1 = w + (size_t)(n0 + 16 + l15) * CC;
    const float* wp2 = w + (size_t)(n0 + 32 + l15) * CC;

    v8f acc0 = {0.f,0.f,0.f,0.f,0.f,0.f,0.f,0.f};
    v8f acc1 = acc0, acc2 = acc0;

    #pragma unroll
    for (int k0 = 0; k0 < CC; k0 += 4) {
        const int kk = k0 + 2 * half;
        const v2f pv = *(const v2f*)(p + kk);
        const v2f qv = *(const v2f*)(q + kk);
        const v2f av = pv * s1 + qv * s2;          // branchless A fragment
        const v2f b0 = *(const v2f*)(wp0 + kk);    // B[k][n] = w[n*C + k]
        const v2f b1 = *(const v2f*)(wp1 + kk);
        const v2f b2 = *(const v2f*)(wp2 + kk);
        acc0 = __builtin_amdgcn_wmma_f32_16x16x4_f32(
                   false, av, false, b0, (short)0, acc0, false, false);
        acc1 = __builtin_amdgcn_wmma_f32_16x16x4_f32(
                   false, av, false, b1, (short)0, acc1, false, false);
        acc2 = __builtin_amdgcn_wmma_f32_16x16x4_f32(
                   false, av, false, b2, (short)0, acc2, false, false);
    }

    const float bv0 = bias[n0 + l15];
    const float bv1 = bias[n0 + 16 + l15];
    const float bv2 = bias[n0 + 32 + l15];
    float* ob = out + (size_t)(s * BB + b) * ((size_t)ROWS * EMB);
    #pragma unroll
    for (int i = 0; i < 8; ++i) {           // D: VGPR i -> M = i + 8*half
        const int r = m0 + half * 8 + i;
        if (r < ROWS) {
            float* rp = ob + (size_t)r * EMB + n0 + l15;
            rp[0]  = acc0[i] + bv0;
            rp[16] = acc1[i] + bv1;
            rp[32] = acc2[i] + bv2;
        }
    }
}

// ---------------------------------------------------------------------------
extern "C" void kernel_launch(void* const* d_in, const int* in_sizes, int n_in,
                              void* d_out, int out_size, void* d_ws, size_t ws_size,
                              hipStream_t stream) {
    const float* x        = (const float*)d_in[0];
    const int*   seg      = (const int*)  d_in[1];
    const float* cls_tok  = (const float*)d_in[2];
    const float* cls_pos  = (const float*)d_in[3];
    const float* w        = (const float*)d_in[4];
    const float* bias     = (const float*)d_in[5];
    float*       out      = (float*)d_out;

    float* sums = (float*)d_ws;                                  // [4][8192][128]
    float* cnts = sums + (size_t)NSC * BB * NSEG_ * CC;          // [4][8192]

    const int nzero = NSC * BB * NSEG_ * CC + NSC * BB * NSEG_;  // 4,227,072
    zero_ws<<<1024, 256, 0, stream>>>(sums, nzero);

    dim3 gr(4, 16, 8);                       // pixel tiles x channel chunks x batch
    seg_reduce<<<gr, 256, 36864 * sizeof(float), stream>>>(x, seg, sums, cnts);

    token_gemm<<<1040, 256, 0, stream>>>(sums, cnts, cls_tok, cls_pos, w, bias, out);

    // seg_out == input segments (identity remap): raw copy into output tail.
    const size_t tokElems = (size_t)NSC * BB * ROWS * EMB;       // 6,297,600
    hipMemcpyAsync(out + tokElems, seg, (size_t)BB * NSC * HWP * sizeof(int),
                   hipMemcpyDeviceToDevice, stream);
}